// LSTM_12635793785542
// MI455X (gfx1250) — compile-verified
//
#include <hip/hip_runtime.h>

typedef __attribute__((ext_vector_type(2))) float v2f;
typedef __attribute__((ext_vector_type(8))) float v8f;

#define TT    48   // timesteps
#define NIN   5    // input features
#define WAVES 4
#define BT    64   // batch rows per block (16 per wave)

// D = A(16x4 f32) * B(4x16 f32) + C(16x16 f32), full f32 precision
#define WMMA4(A, B, C) \
  __builtin_amdgcn_wmma_f32_16x16x4_f32(false, (A), false, (B), (short)0, (C), false, false)

#define LOG2E 1.442695040888963f

// sigmoid(x) = rcp(1 + 2^(-x*log2e)) : v_mul + v_exp + v_add + v_rcp
__device__ __forceinline__ float fsig(float x) {
  return __builtin_amdgcn_rcpf(1.0f + __builtin_amdgcn_exp2f(-LOG2E * x));
}
#if __has_builtin(__builtin_amdgcn_tanhf)
__device__ __forceinline__ float ftanh(float x) { return __builtin_amdgcn_tanhf(x); }
#else
__device__ __forceinline__ float ftanh(float x) { return 2.0f * fsig(2.0f * x) - 1.0f; }
#endif

// Cross-lane XOR exchange within a 16-lane row via DPP16 ROW_XMASK (pure VALU,
// keeps the LDS pipe / DScnt waits off the recurrence critical path).
template <int XMASK>
__device__ __forceinline__ float dpp_xor(float x) {
  return __int_as_float(__builtin_amdgcn_update_dpp(
      0, __float_as_int(x), 0x160 | XMASK, 0xF, 0xF, true));
}

__global__ __launch_bounds__(128) void lstm_wmma_kernel(
    const float* __restrict__ x,     // [B, T, 5]
    const float* __restrict__ W_ih,  // [32, 5]
    const float* __restrict__ W_hh,  // [32, 8]
    const float* __restrict__ b_ih,  // [32]
    const float* __restrict__ b_hh,  // [32]
    const float* __restrict__ fc_W,  // [1, 384]
    const float* __restrict__ fc_b,  // [1]
    float* __restrict__ out)         // [B]
{
  __shared__ __align__(16) float xs[BT][TT * NIN];   // 61440 B staged x
  // h round-trip, transposed [j][m]; row 8 = write dump for lanes n0>=8
  __shared__ __align__(16) float hbT[WAVES][9][16];  // 2304 B
  __shared__ float fcw[TT * 8];                      // 1536 B

  const int tid  = threadIdx.x;
  const int wave = tid >> 5;
  const int lane = tid & 31;
  const int n0   = lane & 15;   // N column within tile / A row m
  const int grp  = lane >> 4;   // lane half
  const int bg0  = blockIdx.x * BT;

  // ---- stage x for this block (coalesced float4 copy: 64 rows x 960B) ----
  {
    const float4* src = (const float4*)(x + (size_t)bg0 * (TT * NIN));
    float4* dst = (float4*)&xs[0][0];
    for (int i = tid; i < BT * TT * NIN / 4; i += 128) dst[i] = src[i];
  }
  for (int i = tid; i < TT * 8; i += 128) fcw[i] = fc_W[i];
  for (int i = tid; i < WAVES * 9 * 16; i += 128) (&hbT[0][0][0])[i] = 0.0f;

  // ---- B-operand tiles (K=4 x N=16): row K = v + 2*grp, col N = 16*nt + n0 ----
  v2f wih[2][2], whh[2][2];
#pragma unroll
  for (int kc = 0; kc < 2; ++kc)
#pragma unroll
    for (int nt = 0; nt < 2; ++nt) {
      const int g  = 16 * nt + n0;       // gate index (row of W_ih/W_hh)
      const int k0 = 4 * kc + 0 + 2 * grp;
      const int k1 = 4 * kc + 1 + 2 * grp;
      v2f wi, wh;
      wi.x = (k0 < NIN) ? W_ih[g * NIN + k0] : 0.0f;   // pad I=5 -> K=8
      wi.y = (k1 < NIN) ? W_ih[g * NIN + k1] : 0.0f;
      wh.x = W_hh[g * 8 + k0];
      wh.y = W_hh[g * 8 + k1];
      wih[kc][nt] = wi;
      whh[kc][nt] = wh;
    }
  const float bias0 = b_ih[n0] + b_hh[n0];
  const float bias1 = b_ih[16 + n0] + b_hh[16 + n0];
  const float fcb   = fc_b[0];

  const int row = wave * 16 + n0;   // this lane's A-tile batch row (within block)

  // h read base: ah0 = {hbT[w][2g][n0], hbT[w][2g+1][n0]}, ah1 rows +4
  const float* hread = &hbT[wave][2 * grp][n0];
  // h write base: rows m = 8*grp..8*grp+7, col j = n0 (dump row 8 if n0>=8)
  float4* hstore = (float4*)&hbT[wave][(n0 < 8) ? n0 : 8][8 * grp];

  v8f cst = {};   // cell state: rows m = v + 8*grp, hidden j = n0 (valid lanes n0<8)
  v8f acc = {};   // fused FC accumulator

  __syncthreads();

  for (int t = 0; t < TT; ++t) {
    // ---- A tiles (16x4 f32): per lane K = v + 2*grp ----
    v2f ax0, ax1, ah0, ah1;
    ax0.x = xs[row][t * NIN + 2 * grp + 0];
    ax0.y = xs[row][t * NIN + 2 * grp + 1];
    const float x4 = xs[row][t * NIN + 4];             // unconditional load
    ax1.x = (grp == 0) ? x4 : 0.0f;                    // K=4 valid, K=5..7 zero
    ax1.y = 0.0f;
    ah0.x = hread[0];
    ah0.y = hread[16];
    ah1.x = hread[64];
    ah1.y = hread[80];

    // ---- gates = bias + x@W_ih^T + h@W_hh^T via 8 chained f32 WMMAs ----
    v8f c0, c1;
#pragma unroll
    for (int v = 0; v < 8; ++v) { c0[v] = bias0; c1[v] = bias1; }
    c0 = WMMA4(ax0, wih[0][0], c0);
    c0 = WMMA4(ax1, wih[1][0], c0);
    c0 = WMMA4(ah0, whh[0][0], c0);
    c0 = WMMA4(ah1, whh[1][0], c0);
    c1 = WMMA4(ax0, wih[0][1], c1);
    c1 = WMMA4(ax1, wih[1][1], c1);
    c1 = WMMA4(ah0, whh[0][1], c1);
    c1 = WMMA4(ah1, whh[1][1], c1);

    // ---- elementwise LSTM cell; c0 = [i|f], c1 = [g|o] split at n=8 ----
    // All lanes compute (lanes n0>=8 produce harmless garbage that lands in
    // the dump row); no EXEC manipulation anywhere in the loop.
    const float wfc = fcw[t * 8 + (n0 & 7)];
    float hv[8];
#pragma unroll
    for (int v = 0; v < 8; ++v) {
      const float a0 = c0[v];
      const float a1 = c1[v];
      const float s0 = fsig(a0);           // i-sig (low lanes) / f-sig (high)
      const float fg = dpp_xor<8>(s0);     // partner's sigmoid -> f on low lanes
      const float p1 = dpp_xor<8>(a1);     // partner's raw g/o preact
      const float gg = ftanh(a1);
      const float og = fsig(p1);
      const float cn = fmaf(fg, cst[v], s0 * gg);
      cst[v] = cn;
      const float hn = og * ftanh(cn);
      acc[v] = fmaf(hn, wfc, acc[v]);      // fused FC: idx = t*H + j
      hv[v] = hn;
    }
    // transpose h back via LDS: two b128 stores, unpredicated
    float4 q0, q1;
    q0.x = hv[0]; q0.y = hv[1]; q0.z = hv[2]; q0.w = hv[3];
    q1.x = hv[4]; q1.y = hv[5]; q1.z = hv[6]; q1.w = hv[7];
    hstore[0] = q0;
    hstore[1] = q1;
    // no barrier: each wave only touches its own LDS region, and per-wave
    // LDS ordering is in-order (DScnt) per the CDNA5 ISA.
  }

  // ---- reduce acc over hidden dim j = lanes 0..7 (per lane-half) ----
#pragma unroll
  for (int v = 0; v < 8; ++v) {
    float a = acc[v];
    a += dpp_xor<1>(a);
    a += dpp_xor<2>(a);
    a += dpp_xor<4>(a);
    acc[v] = a;
  }
  if (n0 == 0) {
#pragma unroll
    for (int v = 0; v < 8; ++v)
      out[bg0 + wave * 16 + 8 * grp + v] = acc[v] + fcb;
  }
}

extern "C" void kernel_launch(void* const* d_in, const int* in_sizes, int n_in,
                              void* d_out, int out_size, void* d_ws, size_t ws_size,
                              hipStream_t stream) {
  (void)n_in; (void)d_ws; (void)ws_size; (void)in_sizes;
  const float* x    = (const float*)d_in[0];
  const float* W_ih = (const float*)d_in[1];
  const float* W_hh = (const float*)d_in[2];
  const float* b_ih = (const float*)d_in[3];
  const float* b_hh = (const float*)d_in[4];
  const float* fc_W = (const float*)d_in[5];
  const float* fc_b = (const float*)d_in[6];
  float* out = (float*)d_out;

  const int B = out_size;              // 131072, divisible by BT=64
  const int blocks = B / BT;
  lstm_wmma_kernel<<<blocks, 128, 0, stream>>>(x, W_ih, W_hh, b_ih, b_hh, fc_W, fc_b, out);
}